// RNN_26388279067168
// MI455X (gfx1250) — compile-verified
//
#include <hip/hip_runtime.h>

#define INPUT_SZ  28
#define HIDDEN    64
#define G4        256          // 4*HIDDEN (i,f,g,o)
#define T_STEPS   28
#define CLASSES   10

#define WAVES_PER_BLOCK 4
#define ROWS_PER_WAVE   16
#define ROWS_PER_BLOCK  (WAVES_PER_BLOCK * ROWS_PER_WAVE)   // 64

#define WIH_STRIDE 28          // floats per LDS row of W_ih (stride 28 is bank-clean)
#define WHH_STRIDE 66          // padded 64->66 to avoid 64-dword bank aliasing
#define H_STRIDE   68          // padded h-tile row stride (4m+{0..3} covers all banks)

typedef __attribute__((ext_vector_type(2))) float v2f;
typedef __attribute__((ext_vector_type(8))) float v8f;

// CDNA5 has a hardware V_TANH_F32 (TRANS32 op). Prefer it; fall back to the
// v_exp/v_rcp formulation if this toolchain lacks the builtin.
__device__ __forceinline__ float hw_tanh(float x) {
#if __has_builtin(__builtin_amdgcn_tanhf)
    return __builtin_amdgcn_tanhf(x);
#elif __has_builtin(__builtin_amdgcn_tanh_f32)
    return __builtin_amdgcn_tanh_f32(x);
#else
    float e = __expf(2.0f * x);
    return 1.0f - 2.0f * __builtin_amdgcn_rcpf(e + 1.0f);
#endif
}
__device__ __forceinline__ float hw_sigmoid(float x) {
    // sigmoid(x) = 0.5*tanh(x/2) + 0.5  -> single trans op on CDNA5
    return __builtin_fmaf(0.5f, hw_tanh(0.5f * x), 0.5f);
}

__global__ __launch_bounds__(WAVES_PER_BLOCK * 32, 1)
void lstm_fused_wmma(const float* __restrict__ x,
                     const float* __restrict__ W_ih,
                     const float* __restrict__ W_hh,
                     const float* __restrict__ b_ih,
                     const float* __restrict__ b_hh,
                     const float* __restrict__ W_cls,
                     const float* __restrict__ b_cls,
                     float* __restrict__ out)
{
    __shared__ float sWih[G4 * WIH_STRIDE];                         // 28.0 KB
    __shared__ float sWhh[G4 * WHH_STRIDE];                         // 66.0 KB
    __shared__ float sBias[G4];                                     //  1.0 KB
    __shared__ float sH[WAVES_PER_BLOCK][ROWS_PER_WAVE * H_STRIDE]; // 17.0 KB

    const int tid  = threadIdx.x;
    const int wave = tid >> 5;
    const int lane = tid & 31;
    const int m    = lane & 15;   // row index (A/D) or column index (B/D) role
    const int hi   = lane >> 4;   // K-half selector for f32 WMMA fragments

    // ---- cooperative staging of weights into LDS ----
    for (int i = tid; i < G4 * INPUT_SZ; i += blockDim.x) sWih[i] = W_ih[i];
    for (int i = tid; i < G4 * HIDDEN; i += blockDim.x) {
        int r = i / HIDDEN, c = i - r * HIDDEN;
        sWhh[r * WHH_STRIDE + c] = W_hh[i];
    }
    for (int i = tid; i < G4; i += blockDim.x) sBias[i] = b_ih[i] + b_hh[i];

    // zero this wave's private h slab (h0 = 0)
    float* hSlab = &sH[wave][0];
    for (int i = lane; i < ROWS_PER_WAVE * H_STRIDE; i += 32) hSlab[i] = 0.0f;
    __syncthreads();

    const int    rowBase = blockIdx.x * ROWS_PER_BLOCK + wave * ROWS_PER_WAVE;
    const float* xRow    = x + (size_t)(rowBase + m) * (T_STEPS * INPUT_SZ);

    // per-column combined bias, hoisted into registers (loop-invariant)
    float bcol[16];
    #pragma unroll
    for (int nt = 0; nt < 16; ++nt) bcol[nt] = sBias[nt * 16 + m];

    // cell state for 16 rows x 64 hidden, striped like C/D tiles: creg[jt][r]
    v8f creg[4];
    {
        v8f z = {0.f,0.f,0.f,0.f,0.f,0.f,0.f,0.f};
        creg[0] = z; creg[1] = z; creg[2] = z; creg[3] = z;
    }

    for (int t = 0; t < T_STEPS; ++t) {
        // gates = (b_ih + b_hh) broadcast per column
        v8f acc[16];
        #pragma unroll
        for (int nt = 0; nt < 16; ++nt) {
            float bc = bcol[nt];
            v8f v = {bc, bc, bc, bc, bc, bc, bc, bc};
            acc[nt] = v;
        }

        // ---- gates += x_t @ W_ih^T   (K = 28 -> 7 WMMA K-steps) ----
        const float* xT = xRow + t * INPUT_SZ;
        #pragma unroll 1
        for (int kk = 0; kk < 7; ++kk) {
            const int k0 = kk * 4 + 2 * hi;
            v2f a = *(const v2f*)(xT + k0);            // 8B-aligned global b64
            #pragma unroll
            for (int nt = 0; nt < 16; ++nt) {
                v2f b = *(const v2f*)&sWih[(nt * 16 + m) * WIH_STRIDE + k0];
                acc[nt] = __builtin_amdgcn_wmma_f32_16x16x4_f32(
                    false, a, false, b, (short)0, acc[nt], false, false);
            }
        }

        // ---- gates += h @ W_hh^T     (K = 64 -> 16 WMMA K-steps) ----
        #pragma unroll 1
        for (int kk = 0; kk < 16; ++kk) {
            const int k0 = kk * 4 + 2 * hi;
            v2f a = *(const v2f*)&hSlab[m * H_STRIDE + k0];
            #pragma unroll
            for (int nt = 0; nt < 16; ++nt) {
                v2f b = *(const v2f*)&sWhh[(nt * 16 + m) * WHH_STRIDE + k0];
                acc[nt] = __builtin_amdgcn_wmma_f32_16x16x4_f32(
                    false, a, false, b, (short)0, acc[nt], false, false);
            }
        }

        // prefetch next timestep's activations while we do elementwise work
        if (t + 1 < T_STEPS) __builtin_prefetch(xT + INPUT_SZ, 0, 1);

        // ---- fused i,f,g,o nonlinearity + cell/hidden update ----
        // lane holds (row = r + 8*hi, col = jt*16 + m) for tiles jt, 4+jt, 8+jt, 12+jt
        #pragma unroll
        for (int jt = 0; jt < 4; ++jt) {
            #pragma unroll
            for (int r = 0; r < 8; ++r) {
                float iv = hw_sigmoid(acc[jt][r]);
                float fv = hw_sigmoid(acc[4 + jt][r]);
                float gv = hw_tanh   (acc[8 + jt][r]);
                float ov = hw_sigmoid(acc[12 + jt][r]);
                float cc = __builtin_fmaf(fv, creg[jt][r], iv * gv);
                creg[jt][r] = cc;
                float hh = ov * hw_tanh(cc);
                hSlab[(r + 8 * hi) * H_STRIDE + jt * 16 + m] = hh;
                // same-wave DS ops are ordered: next step's A-fragment loads see this
            }
        }
    }

    // ---- classifier: out = h_T @ W_cls^T + b_cls (N padded 10 -> 16) ----
    const int   colC = (m < CLASSES) ? m : (CLASSES - 1);  // clamped, in-bounds
    const float bRaw = b_cls[colC];
    const float bc   = (m < CLASSES) ? bRaw : 0.0f;        // v_cndmask, EXEC untouched
    v8f accO = {bc, bc, bc, bc, bc, bc, bc, bc};

    #pragma unroll 1
    for (int kk = 0; kk < 16; ++kk) {
        const int k0 = kk * 4 + 2 * hi;
        v2f a = *(const v2f*)&hSlab[m * H_STRIDE + k0];
        v2f b = *(const v2f*)(W_cls + colC * HIDDEN + k0);
        b.x = (m < CLASSES) ? b.x : 0.0f;                  // branchless pad column
        b.y = (m < CLASSES) ? b.y : 0.0f;
        accO = __builtin_amdgcn_wmma_f32_16x16x4_f32(
            false, a, false, b, (short)0, accO, false, false);
    }

    // store: lane holds out[rowBase + r + 8*hi][m] for m < 10
    if (m < CLASSES) {
        #pragma unroll
        for (int r = 0; r < 8; ++r) {
            out[(size_t)(rowBase + r + 8 * hi) * CLASSES + m] = accO[r];
        }
    }
}

extern "C" void kernel_launch(void* const* d_in, const int* in_sizes, int n_in,
                              void* d_out, int out_size, void* d_ws, size_t ws_size,
                              hipStream_t stream) {
    const float* x     = (const float*)d_in[0];
    const float* W_ih  = (const float*)d_in[1];
    const float* W_hh  = (const float*)d_in[2];
    const float* b_ih  = (const float*)d_in[3];
    const float* b_hh  = (const float*)d_in[4];
    const float* W_cls = (const float*)d_in[5];
    const float* b_cls = (const float*)d_in[6];
    float* out = (float*)d_out;

    const int B = in_sizes[0] / (T_STEPS * INPUT_SZ);      // 16384
    dim3 grid(B / ROWS_PER_BLOCK);                         // 256 workgroups
    dim3 block(WAVES_PER_BLOCK * 32);                      // 4 waves (wave32)
    lstm_fused_wmma<<<grid, block, 0, stream>>>(x, W_ih, W_hh, b_ih, b_hh,
                                                W_cls, b_cls, out);
}